// RelPartialLearnableMultiHeadAttn_8624294330672
// MI455X (gfx1250) — compile-verified
//
#include <hip/hip_runtime.h>
#include <hip/hip_bf16.h>
#include <math.h>
#include <stdint.h>

#define QLEN    1024
#define MEMLEN  1024
#define KLEN    2048
#define BSZ     2
#define NHEAD   16
#define DHEAD   64
#define DMODEL  1024
#define QKV3    3072
#define SCALE   0.125f          /* 1/sqrt(64) */
#define NEGINF  (-1e30f)

typedef __attribute__((ext_vector_type(2))) float v2f;
typedef __attribute__((ext_vector_type(8))) float v8f;

static __device__ __forceinline__ v8f wmma4(v2f a, v2f b, v8f c) {
  // D = A(16x4 f32) * B(4x16 f32) + C(16x16 f32) on the CDNA5 matrix pipe
  return __builtin_amdgcn_wmma_f32_16x16x4_f32(
      /*neg_a=*/false, a, /*neg_b=*/false, b,
      /*c_mod=*/(short)0, c, /*reuse_a=*/false, /*reuse_b=*/false);
}

// async global->LDS copy, GVS addressing: copies from the global address saddr plus voffset to the LDS address in vdst
static __device__ __forceinline__ void async_copy_b128(unsigned lds_addr,
                                                       unsigned voff,
                                                       unsigned long long sbase) {
  asm volatile("global_load_async_to_lds_b128 %0, %1, %2"
               :: "v"(lds_addr), "v"(voff), "s"(sbase) : "memory");
}
static __device__ __forceinline__ void async_copy_b32(unsigned lds_addr,
                                                      unsigned voff,
                                                      unsigned long long sbase) {
  asm volatile("global_load_async_to_lds_b32 %0, %1, %2"
               :: "v"(lds_addr), "v"(voff), "s"(sbase) : "memory");
}
static __device__ __forceinline__ void wait_async0() {
  asm volatile("s_wait_asynccnt 0x0" ::: "memory");
}

#define APITCH 20   /* 80B rows: keeps 16B alignment for b128 async stores */
#define BPITCH 17

// ---------------------------------------------------------------------------
// Generic fp32 WMMA GEMM: C[M,N] = A[M,K] @ B[K,N]  (row-major, lda=K, ldb=ldc=N)
// Block = 128 threads (4 waves). Block tile 64x64, each wave owns 16 rows x 64
// cols. K streamed in 16-wide chunks, double-buffered in LDS via async
// global->LDS DMA (ASYNCcnt); B is transposed by the async engine's per-lane
// LDS scatter.
// ---------------------------------------------------------------------------
__global__ __launch_bounds__(128)
void gemm_f32_wmma(const float* __restrict__ A, const float* __restrict__ B,
                   float* __restrict__ C, int M, int N, int K) {
  __shared__ __attribute__((aligned(16))) float sA[2][64 * APITCH];
  __shared__ __attribute__((aligned(16))) float sB[2][64 * BPITCH];

  const int tid  = threadIdx.x;
  const int lane = tid & 31;
  const int wave = tid >> 5;
  const int half = lane >> 4;     // 0: K pair {0,1}; 1: K pair {2,3}
  const int l16  = lane & 15;
  const int m0   = blockIdx.y * 64;
  const int n0   = blockIdx.x * 64;

  const unsigned sAbase = (unsigned)(uintptr_t)&sA[0][0];
  const unsigned sBbase = (unsigned)(uintptr_t)&sB[0][0];
  const unsigned long long Ab = (unsigned long long)A;
  const unsigned long long Bb = (unsigned long long)B;

  // per-thread staging coordinates (fixed across chunks)
  const int am = tid >> 1, aq = tid & 1;          // A: 2x b128 per thread
  const int bn = tid & 63, bk0 = tid >> 6;        // B: 8x b32 per thread

  // issue one chunk's async copies into buffer `buf` for K-offset k0
  auto stage = [&](int buf, int k0) {
    // A tile 64x16 as 256 b128 transfers (row am, quads aq and aq+2)
#pragma unroll
    for (int p = 0; p < 2; ++p) {
      const int q = aq + p * 2;
      async_copy_b128(sAbase + (unsigned)(buf * 64 * APITCH + am * APITCH + q * 4) * 4u,
                      (unsigned)(((m0 + am) * K + k0 + q * 4) * 4), Ab);
    }
    // B tile 16x64 scattered transposed: row k0+k, column n0+n of B lands in row n, column k of sB
#pragma unroll
    for (int p = 0; p < 8; ++p) {
      const int bk = bk0 + p * 2;
      async_copy_b32(sBbase + (unsigned)(buf * 64 * BPITCH + bn * BPITCH + bk) * 4u,
                     (unsigned)(((k0 + bk) * N + n0 + bn) * 4), Bb);
    }
  };

  v8f acc[4] = {};                // 16 rows x 64 cols per wave

  stage(0, 0);
  wait_async0();
  __syncthreads();

  for (int k0 = 0; k0 < K; k0 += 16) {
    const int cur = (k0 >> 4) & 1;
    if (k0 + 16 < K) stage(1 - cur, k0 + 16);   // overlap DMA with WMMA

#pragma unroll
    for (int kk = 0; kk < 4; ++kk) {
      const int kb = kk * 4 + half * 2;
      v2f af;
      af.x = sA[cur][(wave * 16 + l16) * APITCH + kb];
      af.y = sA[cur][(wave * 16 + l16) * APITCH + kb + 1];
#pragma unroll
      for (int nt = 0; nt < 4; ++nt) {
        v2f bf;
        bf.x = sB[cur][(nt * 16 + l16) * BPITCH + kb];
        bf.y = sB[cur][(nt * 16 + l16) * BPITCH + kb + 1];
        acc[nt] = wmma4(af, bf, acc[nt]);
      }
    }
    wait_async0();
    __syncthreads();
  }

  // store: lane owns column l16 of each 16-col subtile, rows g + half*8
#pragma unroll
  for (int nt = 0; nt < 4; ++nt) {
#pragma unroll
    for (int g = 0; g < 8; ++g) {
      int row = m0 + wave * 16 + g + half * 8;
      int col = n0 + nt * 16 + l16;
      C[(size_t)row * N + col] = acc[nt][g];
    }
  }
}

// ---------------------------------------------------------------------------
// Fused rel-attention: per wave handles (b, head, 16 query rows), flash-style
// online softmax over all key tiles. rel_shift folded into a skewed LDS read:
//   score[i,j] = (AC[i,j] + BDraw[i, j-i+QLEN-1]) * scale   for j <= i+MEMLEN
// (the rel_shift wrap region coincides exactly with the masked region).
// ---------------------------------------------------------------------------
__global__ __launch_bounds__(128)
void attn_fused(const float* __restrict__ wheads,  // [KLEN*BSZ][3072]
                const float* __restrict__ rk,      // [KLEN][1024]
                const float* __restrict__ rwb,     // [16*64]
                const float* __restrict__ rrb,     // [16*64]
                float* __restrict__ attn_vec) {    // [QLEN*BSZ][1024]
  const int tid  = threadIdx.x;
  const int lane = tid & 31;
  const int wave = tid >> 5;
  const int half = lane >> 4;
  const int l16  = lane & 15;
  const int n    = blockIdx.y;            // head
  const int b    = blockIdx.z;            // batch
  const int i0   = (blockIdx.x * 4 + wave) * 16;

  __shared__ float sAC[4][16][17];
  __shared__ float sBD[4][16][33];        // 32 BDraw columns + pad (skew read)
  __shared__ float sP [4][16][17];
  __shared__ float sM[4][16], sL[4][16], sCorr[4][16];

  const float* qrow  = wheads + ((size_t)(MEMLEN + i0 + l16) * BSZ + b) * QKV3 + n * DHEAD;
  const float* kbase = wheads + (size_t)b * QKV3 + DMODEL     + n * DHEAD;
  const float* vbase = wheads + (size_t)b * QKV3 + 2 * DMODEL + n * DHEAD;
  const float* rkn   = rk + n * DHEAD;

  // Q fragments (A-matrix layout): lane = row, half selects K pair within k-step of 4
  v2f qw[16], qr[16];
#pragma unroll
  for (int t = 0; t < 16; ++t) {
    const int c  = t * 4 + half * 2;
    const float x0 = qrow[c], x1 = qrow[c + 1];
    qw[t].x = x0 + rwb[n * DHEAD + c];     qw[t].y = x1 + rwb[n * DHEAD + c + 1];
    qr[t].x = x0 + rrb[n * DHEAD + c];     qr[t].y = x1 + rrb[n * DHEAD + c + 1];
  }

  v8f o[4] = {};
  if (lane < 16) { sM[wave][lane] = -INFINITY; sL[wave][lane] = 0.0f; }
  __builtin_amdgcn_wave_barrier();

  const int jlast = i0 + MEMLEN;          // last key tile start (causal + memory)
  for (int j0 = 0; j0 <= jlast; j0 += 16) {
    // ---- AC tile: (Q + r_w_bias) @ K^T -----------------------------------
    v8f ac = {};
    {
      const float* kp = kbase + ((size_t)(j0 + l16) * BSZ) * QKV3;
#pragma unroll
      for (int t = 0; t < 16; ++t) {
        const int c = t * 4 + half * 2;
        v2f bf; bf.x = kp[c]; bf.y = kp[c + 1];
        ac = wmma4(qw[t], bf, ac);
      }
    }
    // ---- BDraw tiles: (Q + r_r_bias) @ rk^T, columns c0 .. c0+31 ---------
    const int c0 = j0 - i0 + QLEN - 16;
    v8f bd0 = {}, bd1 = {};
    {
      int r0 = c0 + l16;       if (r0 > KLEN - 1) r0 = KLEN - 1;  // clamped rows
      int r1 = c0 + 16 + l16;  if (r1 > KLEN - 1) r1 = KLEN - 1;  // are masked anyway
      const float* p0 = rkn + (size_t)r0 * DMODEL;
      const float* p1 = rkn + (size_t)r1 * DMODEL;
#pragma unroll
      for (int t = 0; t < 16; ++t) {
        const int c = t * 4 + half * 2;
        v2f b0; b0.x = p0[c]; b0.y = p0[c + 1];
        v2f b1; b1.x = p1[c]; b1.y = p1[c + 1];
        bd0 = wmma4(qr[t], b0, bd0);
        bd1 = wmma4(qr[t], b1, bd1);
      }
    }
    // ---- spill accumulators to LDS ---------------------------------------
#pragma unroll
    for (int g = 0; g < 8; ++g) {
      const int rr = g + half * 8;
      sAC[wave][rr][l16]      = ac[g];
      sBD[wave][rr][l16]      = bd0[g];
      sBD[wave][rr][l16 + 16] = bd1[g];
    }
    __builtin_amdgcn_wave_barrier();

    // ---- online softmax (lanes 0..15, one row each) ----------------------
    if (lane < 16) {
      const int r = lane, i = i0 + r;
      const float mold = sM[wave][r], lold = sL[wave][r];
      float s[16], tmax = -INFINITY;
#pragma unroll
      for (int cc = 0; cc < 16; ++cc) {
        const int j = j0 + cc;
        float v = (j <= i + MEMLEN)
                    ? (sAC[wave][r][cc] + sBD[wave][r][cc + 15 - r]) * SCALE
                    : NEGINF;
        s[cc] = v;
        tmax = fmaxf(tmax, v);
      }
      const float mnew = fmaxf(mold, tmax);
      const float cr   = __expf(mold - mnew);
      float ls = 0.0f;
#pragma unroll
      for (int cc = 0; cc < 16; ++cc) {
        const float p = __expf(s[cc] - mnew);
        sP[wave][r][cc] = p;
        ls += p;
      }
      sM[wave][r]    = mnew;
      sL[wave][r]    = lold * cr + ls;
      sCorr[wave][r] = cr;
    }
    __builtin_amdgcn_wave_barrier();

    // ---- rescale running output ------------------------------------------
#pragma unroll
    for (int g = 0; g < 8; ++g) {
      const float cr = sCorr[wave][g + half * 8];
      o[0][g] *= cr; o[1][g] *= cr; o[2][g] *= cr; o[3][g] *= cr;
    }
    // ---- PV: P(16x16) @ V(16x64) via 16 WMMAs ----------------------------
#pragma unroll
    for (int t = 0; t < 4; ++t) {
      const int kb = t * 4 + half * 2;
      v2f af; af.x = sP[wave][l16][kb]; af.y = sP[wave][l16][kb + 1];
      const float* vp0 = vbase + ((size_t)(j0 + kb)     * BSZ) * QKV3;
      const float* vp1 = vbase + ((size_t)(j0 + kb + 1) * BSZ) * QKV3;
#pragma unroll
      for (int dt = 0; dt < 4; ++dt) {
        const int dc = dt * 16 + l16;
        v2f bf; bf.x = vp0[dc]; bf.y = vp1[dc];
        o[dt] = wmma4(af, bf, o[dt]);
      }
    }
    __builtin_amdgcn_wave_barrier();
  }

  // ---- normalize by softmax denominator and store attn_vec ---------------
  if (lane < 16) sCorr[wave][lane] = 1.0f / sL[wave][lane];
  __builtin_amdgcn_wave_barrier();
#pragma unroll
  for (int g = 0; g < 8; ++g) {
    const int rr = g + half * 8;
    const float inv = sCorr[wave][rr];
    const size_t row = (size_t)(i0 + rr) * BSZ + b;
#pragma unroll
    for (int dt = 0; dt < 4; ++dt)
      attn_vec[row * DMODEL + n * DHEAD + dt * 16 + l16] = o[dt][g] * inv;
  }
}

// ---------------------------------------------------------------------------
// Residual add + LayerNorm, one block per (q, b) row of 1024 elements.
// ---------------------------------------------------------------------------
__global__ __launch_bounds__(256)
void ln_residual(const float* __restrict__ w, const float* __restrict__ aout,
                 const float* __restrict__ gamma, const float* __restrict__ beta,
                 float* __restrict__ out) {
  const int row = blockIdx.x;
  const int tid = threadIdx.x;
  __shared__ float sred[16];

  const float* xw = w    + (size_t)row * DMODEL;
  const float* xa = aout + (size_t)row * DMODEL;
  float x[4], s = 0.0f, sq = 0.0f;
#pragma unroll
  for (int p = 0; p < 4; ++p) {
    const int c = tid + p * 256;
    x[p] = xw[c] + xa[c];
    s += x[p]; sq += x[p] * x[p];
  }
#pragma unroll
  for (int off = 16; off > 0; off >>= 1) {
    s  += __shfl_xor(s,  off, 32);
    sq += __shfl_xor(sq, off, 32);
  }
  if ((tid & 31) == 0) { sred[tid >> 5] = s; sred[8 + (tid >> 5)] = sq; }
  __syncthreads();
  float ts = 0.0f, tq = 0.0f;
#pragma unroll
  for (int k = 0; k < 8; ++k) { ts += sred[k]; tq += sred[8 + k]; }
  const float mu  = ts * (1.0f / DMODEL);
  const float var = tq * (1.0f / DMODEL) - mu * mu;
  const float inv = rsqrtf(var + 1e-5f);
#pragma unroll
  for (int p = 0; p < 4; ++p) {
    const int c = tid + p * 256;
    out[(size_t)row * DMODEL + c] = (x[p] - mu) * inv * gamma[c] + beta[c];
  }
}

// ---------------------------------------------------------------------------
extern "C" void kernel_launch(void* const* d_in, const int* in_sizes, int n_in,
                              void* d_out, int out_size, void* d_ws, size_t ws_size,
                              hipStream_t stream) {
  const float* w     = (const float*)d_in[0];   // [1024,2,1024]
  const float* r     = (const float*)d_in[1];   // [2048,1024]
  const float* rwb   = (const float*)d_in[2];   // [16,64]
  const float* rrb   = (const float*)d_in[3];   // [16,64]
  const float* mems  = (const float*)d_in[4];   // [1024,2,1024]
  /* d_in[5] = attn_mask: causal+memlen, reproduced analytically */
  const float* Wqkv  = (const float*)d_in[6];   // [1024,3072]
  const float* Wr    = (const float*)d_in[7];   // [1024,1024]
  const float* Wo    = (const float*)d_in[8];   // [1024,1024]
  const float* gamma = (const float*)d_in[9];
  const float* beta  = (const float*)d_in[10];

  float* ws      = (float*)d_ws;
  float* wheads  = ws;                                     // 4096*3072 (48 MB)
  float* rk      = wheads + (size_t)4096 * QKV3;           // 2048*1024 ( 8 MB)
  float* avec    = rk     + (size_t)KLEN * DMODEL;         // 2048*1024 ( 8 MB)
  float* aoutbuf = avec   + (size_t)QLEN * BSZ * DMODEL;   // 2048*1024 ( 8 MB)

  const dim3 blk(128);
  // QKV projection: cat([mems; w]) @ W_qkv  (two launches, rows are contiguous)
  gemm_f32_wmma<<<dim3(QKV3 / 64, (MEMLEN * BSZ) / 64), blk, 0, stream>>>(
      mems, Wqkv, wheads, MEMLEN * BSZ, QKV3, DMODEL);
  gemm_f32_wmma<<<dim3(QKV3 / 64, (QLEN * BSZ) / 64), blk, 0, stream>>>(
      w, Wqkv, wheads + (size_t)(MEMLEN * BSZ) * QKV3, QLEN * BSZ, QKV3, DMODEL);
  // Relative-position projection: r @ W_r
  gemm_f32_wmma<<<dim3(DMODEL / 64, KLEN / 64), blk, 0, stream>>>(
      r, Wr, rk, KLEN, DMODEL, DMODEL);
  // Fused rel-shift attention + softmax + PV
  attn_fused<<<dim3(QLEN / 64, NHEAD, BSZ), blk, 0, stream>>>(
      wheads, rk, rwb, rrb, avec);
  // Output projection
  gemm_f32_wmma<<<dim3(DMODEL / 64, (QLEN * BSZ) / 64), blk, 0, stream>>>(
      avec, Wo, aoutbuf, QLEN * BSZ, DMODEL, DMODEL);
  // Residual + LayerNorm
  ln_residual<<<dim3(QLEN * BSZ), dim3(256), 0, stream>>>(
      w, aoutbuf, gamma, beta, (float*)d_out);
}